// FocalLoss_11166914970345
// MI455X (gfx1250) — compile-verified
//
#include <hip/hip_runtime.h>
#include <stdint.h>

// ---------------------------------------------------------------------------
// Focal loss (RetinaNet) for MI455X / gfx1250.
// B=8, N=65536, C=80, M=64. Inputs ~177MB: ~7.6us at HBM (23.3TB/s), but the
// working set fits the 192MB L2, so replays can be L2-resident -> trim VALU.
// Pass 1 per anchor: exists-tests for pos/neg (no div, no argmax).
// Pass 2 (pos anchors only, wave-skipped): exact first-occurrence argmax.
// CDNA5 paths: global_load_async_to_lds_b128 (ASYNCcnt), s_wait_asynccnt,
// global_prefetch_b8.
// ---------------------------------------------------------------------------

namespace {
constexpr int   kB = 8;
constexpr int   kN = 65536;
constexpr int   kC = 80;
constexpr int   kM = 64;
constexpr float kAlpha = 0.25f;
constexpr float kEps   = 1e-4f;
constexpr float kBeta  = 1.0f / 9.0f;
}

#ifndef __has_builtin
#define __has_builtin(x) 0
#endif

// CDNA5 async global->LDS copy (tracked by ASYNCcnt). lds_off is the LDS byte
// address (low 32 bits of the flat shared pointer == AS3 offset).
__device__ __forceinline__ void async_load_to_lds_b128(uint32_t lds_off, uint64_t gaddr) {
    asm volatile("global_load_async_to_lds_b128 %0, %1, off"
                 :: "v"(lds_off), "v"(gaddr)
                 : "memory");
}

__device__ __forceinline__ void wait_asynccnt_zero() {
#if __has_builtin(__builtin_amdgcn_s_wait_asynccnt)
    __builtin_amdgcn_s_wait_asynccnt(0);
#else
    asm volatile("s_wait_asynccnt 0" ::: "memory");
#endif
}

__device__ __forceinline__ float smooth_l1(float x) {
    float d = fabsf(x);
    return (d < kBeta) ? (4.5f * d * d) : (d - (0.5f / 9.0f));
}

// Negative-anchor focal term: (1-a) * p^2 * (-log(1-p)). ~6 VALU + 1 TRANS.
__device__ __forceinline__ float neg_term(float p) {
    p = fminf(fmaxf(p, kEps), 1.0f - kEps);
    const float om = 1.0f - p;
    return (p * p) * (__logf(om) * -(1.0f - kAlpha));
}

__global__ __launch_bounds__(256)
void focal_main_kernel(const float* __restrict__ cls,
                       const float* __restrict__ reg,
                       const float* __restrict__ anc,
                       const float* __restrict__ ann,
                       float* __restrict__ acc)   // acc[b*4 + {cls,reg,npos}]
{
    const int b   = blockIdx.y;
    const int tid = threadIdx.x;

    __shared__ __align__(16) float  s_raw[kM * 5];
    __shared__ float4 s_box[kM];
    __shared__ float  s_lab[kM];
    __shared__ float  s_area[kM];
    __shared__ float  s_red[3];

    if (tid < 3) s_red[tid] = 0.0f;

    // Stage this image's 64 annotations (320 floats = 80 x b128) into LDS
    // with the CDNA5 async path.
    if (tid < (kM * 5) / 4) {
        uint64_t g = (uint64_t)(uintptr_t)(ann + (size_t)b * kM * 5 + tid * 4);
        uint32_t l = (uint32_t)(uintptr_t)(&s_raw[tid * 4]);
        async_load_to_lds_b128(l, g);
    }
    wait_asynccnt_zero();
    __syncthreads();

    if (tid < kM) {
        const float x1 = s_raw[tid * 5 + 0];
        const float y1 = s_raw[tid * 5 + 1];
        const float x2 = s_raw[tid * 5 + 2];
        const float y2 = s_raw[tid * 5 + 3];
        s_box[tid]  = make_float4(x1, y1, x2, y2);
        s_lab[tid]  = s_raw[tid * 5 + 4];
        s_area[tid] = (x2 - x1) * (y2 - y1);
    }
    __syncthreads();

    float cls_sum = 0.0f, reg_sum = 0.0f, npos = 0.0f;

    const int per_block = kN / gridDim.x;
    const int base      = blockIdx.x * per_block;

    for (int it = 0; it < per_block; it += blockDim.x) {
        const int n = base + it + tid;

        const float4 a   = reinterpret_cast<const float4*>(anc)[n];
        const float  aw  = a.z - a.x;
        const float  ah  = a.w - a.y;
        const float  acx = a.x + 0.5f * aw;
        const float  acy = a.y + 0.5f * ah;
        const float  area_a = aw * ah;

        const float* cptr = cls + ((size_t)b * kN + n) * kC;
        // Prefetch the 320B classification row (3 cachelines) behind IoU loop.
        __builtin_prefetch(cptr,      0, 1);
        __builtin_prefetch(cptr + 32, 0, 1);
        __builtin_prefetch(cptr + 64, 0, 1);

        // Pass 1: pos/neg via exists-tests, sign-exact vs reference:
        //   pos  <=> max_m (inter - 0.5*ua) > 0
        //   !neg <=> max_m (inter - 0.4*ua) >= 0
        // ua >= max(area_a, area_b) >= 256, so the 1e-8 clamp never binds.
        float m5 = -1.0f;   // running max of inter - 0.5*ua
        float m4 = -1.0f;   // running max of inter - 0.4*ua
        for (int m = 0; m < kM; ++m) {
            if (s_lab[m] < 0.0f) continue;       // wave-uniform (per image)
            const float4 g  = s_box[m];
            float iw = fminf(a.z, g.z) - fmaxf(a.x, g.x);
            float ih = fminf(a.w, g.w) - fmaxf(a.y, g.y);
            iw = fmaxf(iw, 0.0f);
            ih = fmaxf(ih, 0.0f);
            const float inter = iw * ih;
            const float ua    = (area_a + s_area[m]) - inter;
            m5 = fmaxf(m5, fmaf(-0.5f, ua, inter));
            m4 = fmaxf(m4, fmaf(-0.4f, ua, inter));
        }
        const bool pos = m5 > 0.0f;
        const bool neg = m4 < 0.0f;

        if (neg) {
            // Common path: all targets are 0 -> no label compares.
            float s = 0.0f;
            for (int c0 = 0; c0 < kC; c0 += 4) {
                const float4 pv = *reinterpret_cast<const float4*>(cptr + c0);
                s += neg_term(pv.x) + neg_term(pv.y) +
                     neg_term(pv.z) + neg_term(pv.w);
            }
            cls_sum += s;
        } else if (pos) {
            npos += 1.0f;

            // Pass 2 (rare, wave-skips when no lane is positive): exact
            // first-occurrence argmax via cross-multiplication (no divides).
            float bn = -1.0f, bd = 1.0f;
            int   barg = 0;
            for (int m = 0; m < kM; ++m) {
                if (s_lab[m] < 0.0f) continue;
                const float4 g  = s_box[m];
                float iw = fminf(a.z, g.z) - fmaxf(a.x, g.x);
                float ih = fminf(a.w, g.w) - fmaxf(a.y, g.y);
                iw = fmaxf(iw, 0.0f);
                ih = fmaxf(ih, 0.0f);
                const float inter = iw * ih;
                const float ua    = (area_a + s_area[m]) - inter;
                if (inter * bd > bn * ua) { bn = inter; bd = ua; barg = m; }
            }
            const int lab = (int)s_lab[barg];

            float s = 0.0f;
            for (int c0 = 0; c0 < kC; c0 += 4) {
                const float4 pv = *reinterpret_cast<const float4*>(cptr + c0);
                #pragma unroll
                for (int j = 0; j < 4; ++j) {
                    float p = (j == 0) ? pv.x : (j == 1) ? pv.y : (j == 2) ? pv.z : pv.w;
                    p = fminf(fmaxf(p, kEps), 1.0f - kEps);
                    const float om   = 1.0f - p;
                    const bool  is_t = (c0 + j) == lab;
                    const float arg  = is_t ? p : om;
                    const float w    = is_t ? (kAlpha * om * om)
                                            : ((1.0f - kAlpha) * p * p);
                    s += w * (-__logf(arg));
                }
            }
            cls_sum += s;

            // Smooth-L1 regression term.
            const float4 g  = s_box[barg];
            float gw  = g.z - g.x;
            float gh  = g.w - g.y;
            const float gcx = g.x + 0.5f * gw;
            const float gcy = g.y + 0.5f * gh;
            gw = fmaxf(gw, 1.0f);
            gh = fmaxf(gh, 1.0f);
            const float t0 = (gcx - acx) / aw * 10.0f;   // / 0.1
            const float t1 = (gcy - acy) / ah * 10.0f;   // / 0.1
            const float t2 = __logf(gw / aw) * 5.0f;     // / 0.2
            const float t3 = __logf(gh / ah) * 5.0f;     // / 0.2
            const float4 r = *reinterpret_cast<const float4*>(reg + ((size_t)b * kN + n) * 4);
            reg_sum += smooth_l1(t0 - r.x) + smooth_l1(t1 - r.y) +
                       smooth_l1(t2 - r.z) + smooth_l1(t3 - r.w);
        }
        // else: "ignore" anchor (0.4 <= iou <= 0.5): contributes exactly 0,
        // its 320B classification row is never read.
    }

    // Wave32 shuffle reduction, then LDS, then 3 global atomics per block.
    for (int off = 16; off > 0; off >>= 1) {
        cls_sum += __shfl_down(cls_sum, off, 32);
        reg_sum += __shfl_down(reg_sum, off, 32);
        npos    += __shfl_down(npos,    off, 32);
    }
    if ((tid & 31) == 0) {
        atomicAdd(&s_red[0], cls_sum);
        atomicAdd(&s_red[1], reg_sum);
        atomicAdd(&s_red[2], npos);
    }
    __syncthreads();
    if (tid == 0) {
        atomicAdd(&acc[b * 4 + 0], s_red[0]);
        atomicAdd(&acc[b * 4 + 1], s_red[1]);
        atomicAdd(&acc[b * 4 + 2], s_red[2]);
    }
}

__global__ void focal_final_kernel(const float* __restrict__ acc,
                                   float* __restrict__ out) {
    const int t = threadIdx.x;
    float v = 0.0f;
    if (t < kB) {
        const float cs = acc[t * 4 + 0];
        const float rs = acc[t * 4 + 1];
        const float np = acc[t * 4 + 2];
        const float d  = fmaxf(np, 1.0f);
        const float cl = cs / d;
        const float rl = (np > 0.0f) ? (rs / (d * 4.0f)) : 0.0f;
        v = (cl + rl) * (1.0f / (float)kB);
    }
    for (int off = 16; off > 0; off >>= 1) v += __shfl_down(v, off, 32);
    if (t == 0) out[0] = v;
}

extern "C" void kernel_launch(void* const* d_in, const int* in_sizes, int n_in,
                              void* d_out, int out_size, void* d_ws, size_t ws_size,
                              hipStream_t stream) {
    const float* cls = (const float*)d_in[0];   // (8, 65536, 80)
    const float* reg = (const float*)d_in[1];   // (8, 65536, 4)
    const float* anc = (const float*)d_in[2];   // (1, 65536, 4)
    const float* ann = (const float*)d_in[3];   // (8, 64, 5)

    float* acc = (float*)d_ws;                  // 8 images x 4 floats
    hipMemsetAsync(acc, 0, kB * 4 * sizeof(float), stream);

    dim3 grid(64, kB);                          // 1024 anchors per block
    focal_main_kernel<<<grid, 256, 0, stream>>>(cls, reg, anc, ann, acc);
    focal_final_kernel<<<1, 32, 0, stream>>>(acc, (float*)d_out);
}